// NAMTM_55138790146748
// MI455X (gfx1250) — compile-verified
//
#include <hip/hip_runtime.h>
#include <math.h>

// ---------------------------------------------------------------------------
// Soft-tape (NTM-style) recurrence, chunked delta-rule formulation for CDNA5.
//   values: [S,B,D] f32, Wd:[D,3], bd:[3], Wrw:[D,2], brw:[2]
//   outputs: reads [S,B,D] then tape [T,B,D], concatenated in d_out.
// ---------------------------------------------------------------------------

#define S_LEN 512
#define B_N   128
#define D_DIM 512
#define T_SZ  64
#define CHUNK 16
#define NCHUNK (S_LEN / CHUNK)

// padded LDS pitches (bank-conflict avoidance, 64 banks)
#define DP  532   // pitch for tape / ulds / aold rows (D=512 + 20)
#define PCP 69    // pitch for Pc rows (64 + 5)

typedef __attribute__((ext_vector_type(2))) float v2f;
typedef __attribute__((ext_vector_type(8))) float v8f;

// ---------------------------------------------------------------------------
// Kernel A: per-(s,b) controls: softmax(values@Wd+bd) and sigmoid(values@Wrw+brw)
// ctrl[(s*B+b)*5 + {0,1,2}] = directions, {3,4} = r/w probs.
// One wave (32 lanes) per (s,b); shuffle reduction.
// ---------------------------------------------------------------------------
__global__ void NAMTM_ctrl_kernel(const float* __restrict__ values,
                                  const float* __restrict__ Wd,
                                  const float* __restrict__ bd,
                                  const float* __restrict__ Wrw,
                                  const float* __restrict__ brw,
                                  float* __restrict__ ctrl) {
  const int tid  = threadIdx.x;
  const int lane = tid & 31;
  const int wv   = blockIdx.x * (blockDim.x >> 5) + (tid >> 5); // = s*B + b
  if (wv >= S_LEN * B_N) return;

  const float* vp = values + (size_t)wv * D_DIM;
  float a0 = 0.f, a1 = 0.f, a2 = 0.f, a3 = 0.f, a4 = 0.f;
  for (int i = lane; i < D_DIM; i += 32) {
    float x = vp[i];
    a0 += x * Wd[i * 3 + 0];
    a1 += x * Wd[i * 3 + 1];
    a2 += x * Wd[i * 3 + 2];
    a3 += x * Wrw[i * 2 + 0];
    a4 += x * Wrw[i * 2 + 1];
  }
  for (int off = 16; off > 0; off >>= 1) {
    a0 += __shfl_xor(a0, off, 32);
    a1 += __shfl_xor(a1, off, 32);
    a2 += __shfl_xor(a2, off, 32);
    a3 += __shfl_xor(a3, off, 32);
    a4 += __shfl_xor(a4, off, 32);
  }
  if (lane == 0) {
    float l0 = a0 + bd[0], l1 = a1 + bd[1], l2 = a2 + bd[2];
    float mx = fmaxf(l0, fmaxf(l1, l2));
    float e0 = __expf(l0 - mx), e1 = __expf(l1 - mx), e2 = __expf(l2 - mx);
    float inv = 1.0f / (e0 + e1 + e2);
    float* c = ctrl + (size_t)wv * 5;
    c[0] = e0 * inv;
    c[1] = e1 * inv;
    c[2] = e2 * inv;
    c[3] = 1.0f / (1.0f + __expf(-(a3 + brw[0])));
    c[4] = 1.0f / (1.0f + __expf(-(a4 + brw[1])));
  }
}

// ---------------------------------------------------------------------------
// Kernel B: one workgroup per batch element b. Tape state in LDS.
// Chunked delta rule with V_WMMA_F32_16X16X4_F32 for the two matmuls.
// 512 threads = 16 waves (wave32).
// ---------------------------------------------------------------------------
__global__ void __launch_bounds__(512, 1)
NAMTM_tape_kernel(const float* __restrict__ values,
                  const float* __restrict__ ctrl,
                  float* __restrict__ out_reads,
                  float* __restrict__ out_tape) {
  extern __shared__ float smem[];
  float* tape = smem;                       // [T_SZ][DP]
  float* ulds = tape + T_SZ * DP;           // [CHUNK][DP]
  float* aold = ulds + CHUNK * DP;          // [CHUNK][DP]
  float* Pc   = aold + CHUNK * DP;          // [CHUNK][PCP]  (pos per chunk step)
  float* G    = Pc + CHUNK * PCP;           // [CHUNK][CHUNK]
  float* pos  = G + CHUNK * CHUNK;          // [T_SZ]
  float* wch  = pos + T_SZ;                 // [CHUNK] write gates
  float* rch  = wch + CHUNK;                // [CHUNK] read gates

  const int b    = blockIdx.x;
  const int tid  = threadIdx.x;
  const int lane = tid & 31;
  const int wave = tid >> 5;
  const int m    = lane & 15;   // column/row within 16-wide fragment
  const int kh   = lane >> 4;   // lane-half (K split 0/1 vs 2/3, M split 0-7 vs 8-15)

  // init tape = 0, pos = one-hot(0)
  for (int i = tid; i < T_SZ * DP; i += 512) tape[i] = 0.f;
  if (tid < T_SZ) pos[tid] = (tid == 0) ? 1.f : 0.f;
  __syncthreads();

  for (int c = 0; c < NCHUNK; ++c) {
    const int s0 = c * CHUNK;

    // per-chunk read/write gates
    if (tid < CHUNK) {
      const float* cc = ctrl + ((size_t)(s0 + tid) * B_N + b) * 5;
      rch[tid] = cc[3];
      wch[tid] = cc[4];
    }

    // prefetch next chunk's value tile into cache (global_prefetch_b8);
    // overlaps with pos-trajectory + WMMA phases below.
    if (c + 1 < NCHUNK) {
      const size_t pbase = ((size_t)(s0 + CHUNK) * B_N + b) * D_DIM + tid;
#pragma unroll
      for (int s = 0; s < CHUNK; ++s)
        __builtin_prefetch(&values[pbase + (size_t)s * B_N * D_DIM], 0, 3);
    }

    // ---- phase 1: record pos trajectory for this chunk, advance pos --------
    for (int s = 0; s < CHUNK; ++s) {
      if (tid < T_SZ) Pc[s * PCP + tid] = pos[tid];  // own element, no race
      __syncthreads();
      float pn = 0.f;
      if (tid < T_SZ) {
        const float* cc = ctrl + ((size_t)(s0 + s) * B_N + b) * 5;
        float d0 = cc[0], d1 = cc[1], d2 = cc[2];
        // prevpos[l]=pos[l+1], nextpos[l]=pos[l-1] (circular, T=64)
        pn = pos[(tid + 1) & 63] * d0 + pos[tid] * d1 + pos[(tid + 63) & 63] * d2;
      }
      __syncthreads();
      if (tid < T_SZ) pos[tid] = pn;
    }
    __syncthreads();

    // ---- phase 2: Gram matrix G = Pc * Pc^T (16x16, tiny) ------------------
    if (tid < CHUNK * CHUNK) {
      int si = tid >> 4, j = tid & 15;
      float g = 0.f;
      for (int l = 0; l < T_SZ; ++l) g += Pc[si * PCP + l] * Pc[j * PCP + l];
      G[tid] = g;
    }
    __syncthreads();

    // ---- phase 3: A = Pc (16x64) x tape (64x512) via WMMA fp32 -------------
    // 32 N-tiles of 16; 16 waves -> 2 tiles each; K=64 in 16 steps of 4.
    // A-fragments hoisted to registers once, shared by both N-tiles; the two
    // accumulator chains are independent, letting the scheduler interleave
    // WMMAs without dead cycles.
    {
      v2f afrag[16];
#pragma unroll
      for (int kk = 0; kk < 16; ++kk) {
        const int kb = 4 * kk + 2 * kh;
        afrag[kk].x = Pc[m * PCP + kb + 0];   // A[M=m][K=kb]
        afrag[kk].y = Pc[m * PCP + kb + 1];   // A[M=m][K=kb+1]
      }
      const int n0a = wave * 16;
      const int n0b = (wave + 16) * 16;
      v8f acc0 = {0.f, 0.f, 0.f, 0.f, 0.f, 0.f, 0.f, 0.f};
      v8f acc1 = {0.f, 0.f, 0.f, 0.f, 0.f, 0.f, 0.f, 0.f};
#pragma unroll
      for (int kk = 0; kk < 16; ++kk) {
        const int kb = 4 * kk + 2 * kh;
        v2f bv0, bv1;
        bv0.x = tape[(kb + 0) * DP + n0a + m];  // B[K=kb][N=m]
        bv0.y = tape[(kb + 1) * DP + n0a + m];
        bv1.x = tape[(kb + 0) * DP + n0b + m];
        bv1.y = tape[(kb + 1) * DP + n0b + m];
        acc0 = __builtin_amdgcn_wmma_f32_16x16x4_f32(
            false, afrag[kk], false, bv0, (short)0, acc0, false, false);
        acc1 = __builtin_amdgcn_wmma_f32_16x16x4_f32(
            false, afrag[kk], false, bv1, (short)0, acc1, false, false);
      }
#pragma unroll
      for (int r = 0; r < 8; ++r) {           // D: lanes 0-15 rows 0-7, 16-31 rows 8-15
        aold[(r + 8 * kh) * DP + n0a + m] = acc0[r];
        aold[(r + 8 * kh) * DP + n0b + m] = acc1[r];
      }
    }
    __syncthreads();

    // ---- phase 4: intra-chunk sequential delta rule (thread owns column d) -
    {
      const int d = tid;  // 512 threads, D=512
      float u[CHUNK];
      const size_t gbase = ((size_t)s0 * B_N + b) * D_DIM + d;
#pragma unroll
      for (int s = 0; s < CHUNK; ++s) {
        float ov = aold[s * DP + d];
#pragma unroll
        for (int j = 0; j < s; ++j) ov += G[s * CHUNK + j] * u[j];
        float vv = values[gbase + (size_t)s * B_N * D_DIM];
        float us = wch[s] * (vv - ov);
        u[s] = us;
        ulds[s * DP + d] = us;
        out_reads[gbase + (size_t)s * B_N * D_DIM] = rch[s] * ov;
      }
    }
    __syncthreads();

    // ---- phase 5: tape (64x512) += Pc^T (64x16) x U (16x512) via WMMA ------
    // 4 M-tiles x 32 N-tiles = 128 tiles; each wave owns N-tiles {wave, wave+16}
    // x all 4 M-tiles. B-fragments (ulds) loaded once per N-tile and reused
    // across the 4 M-tiles; the 4 accumulator chains are independent.
#pragma unroll
    for (int tn = 0; tn < 2; ++tn) {
      const int n0 = (wave + 16 * tn) * 16;
      v2f bfrag[4];
#pragma unroll
      for (int kk = 0; kk < 4; ++kk) {
        const int kb = 4 * kk + 2 * kh;
        bfrag[kk].x = ulds[(kb + 0) * DP + n0 + m];  // B[K=kb][N=m]
        bfrag[kk].y = ulds[(kb + 1) * DP + n0 + m];
      }
#pragma unroll
      for (int mt = 0; mt < 4; ++mt) {
        const int m0 = mt * 16;
        v8f acc;
#pragma unroll
        for (int r = 0; r < 8; ++r)
          acc[r] = tape[(m0 + r + 8 * kh) * DP + n0 + m];
#pragma unroll
        for (int kk = 0; kk < 4; ++kk) {
          const int kb = 4 * kk + 2 * kh;
          v2f av;
          av.x = Pc[(kb + 0) * PCP + m0 + m];   // A[M=m0+m][K=kb] = Pc[kb][m0+m]
          av.y = Pc[(kb + 1) * PCP + m0 + m];
          acc = __builtin_amdgcn_wmma_f32_16x16x4_f32(
              false, av, false, bfrag[kk], (short)0, acc, false, false);
        }
#pragma unroll
        for (int r = 0; r < 8; ++r)
          tape[(m0 + r + 8 * kh) * DP + n0 + m] = acc[r];
      }
    }
    __syncthreads();
  }

  // ---- epilogue: write final tape [T,B,D] ----------------------------------
  for (int i = tid; i < T_SZ * D_DIM; i += 512) {
    int l = i >> 9;        // slot
    int d = i & 511;       // channel
    out_tape[((size_t)l * B_N + b) * D_DIM + d] = tape[l * DP + d];
  }
}

// ---------------------------------------------------------------------------
extern "C" void kernel_launch(void* const* d_in, const int* in_sizes, int n_in,
                              void* d_out, int out_size, void* d_ws, size_t ws_size,
                              hipStream_t stream) {
  (void)in_sizes; (void)n_in; (void)out_size; (void)ws_size;

  const float* values = (const float*)d_in[0];
  const float* Wd     = (const float*)d_in[1];
  const float* bd     = (const float*)d_in[2];
  const float* Wrw    = (const float*)d_in[3];
  const float* brw    = (const float*)d_in[4];

  float* ctrl      = (float*)d_ws;  // S*B*5 floats = 1.31 MB scratch
  float* out_reads = (float*)d_out;
  float* out_tape  = out_reads + (size_t)S_LEN * B_N * D_DIM;

  // controls: one wave per (s,b); 8 waves per 256-thread block
  NAMTM_ctrl_kernel<<<(S_LEN * B_N) / 8, 256, 0, stream>>>(
      values, Wd, bd, Wrw, brw, ctrl);

  const size_t lds_floats =
      (size_t)T_SZ * DP + 2 * (size_t)CHUNK * DP + (size_t)CHUNK * PCP +
      CHUNK * CHUNK + T_SZ + 2 * CHUNK;
  const size_t lds_bytes = lds_floats * sizeof(float);  // ~210 KB (< 320 KB/WGP)

  (void)hipFuncSetAttribute((const void*)NAMTM_tape_kernel,
                            hipFuncAttributeMaxDynamicSharedMemorySize,
                            (int)lds_bytes);

  NAMTM_tape_kernel<<<B_N, 512, lds_bytes, stream>>>(
      values, ctrl, out_reads, out_tape);
}